// TRTEfficientNMS_73538430042611
// MI455X (gfx1250) — compile-verified
//
#include <hip/hip_runtime.h>
#include <cstdint>

// ---------------- problem constants (match reference) ----------------
#define CCLS      80
#define TILE      128          // boxes per phase-1 block (40 KB LDS tile)
#define NMS_BLOCK 1024         // 32 wave32s per block
#define NBLK      32           // cooperating blocks in phase 2
#define NEG_INF   (-1e30f)
#define IOU_THR   0.6f
#define MAX_OUT   100

// =====================================================================
// Phase 1: per-box class max + argmax.
// Full tiles: async-DMA the 128x80 f32 score tile into LDS
// (global_load_async_to_lds_b128, ASYNCcnt), then each lane scans its
// row from LDS. Partial boundary tile: guarded direct loads.
// =====================================================================
__global__ __launch_bounds__(TILE) void nms_phase1_classmax(
    const float* __restrict__ scores,
    float* __restrict__ live,
    int*   __restrict__ labels,
    int n)
{
    __shared__ __align__(16) float stile[TILE * CCLS];   // 40960 bytes

    const int tid       = threadIdx.x;
    const int tile_base = blockIdx.x * TILE;

    if (tile_base + TILE <= n) {
#if defined(__HIP_DEVICE_COMPILE__)
        const unsigned long long gbase =
            (unsigned long long)(scores + (size_t)tile_base * CCLS);
        const unsigned lds0 =
            (unsigned)(unsigned long long)
            (__attribute__((address_space(3))) char*)&stile[0];

        const int tile_bytes = TILE * CCLS * 4;          // 40960
        for (int k = 0; k < tile_bytes; k += TILE * 16) {
            unsigned goff = (unsigned)(k + tid * 16);
            unsigned ldsa = lds0 + goff;
            asm volatile(
                "global_load_async_to_lds_b128 %0, %1, %2"
                :
                : "v"(ldsa), "v"(goff), "s"(gbase)
                : "memory");
        }
        asm volatile("s_wait_asynccnt 0" ::: "memory");
        __syncthreads();
#else
        for (int k = tid; k < TILE * CCLS; k += TILE)
            stile[k] = scores[(size_t)tile_base * CCLS + k];
        __syncthreads();
#endif
        float m = -3.0e38f; int lab = 0;
        const float* row = &stile[tid * CCLS];
        #pragma unroll 8
        for (int c = 0; c < CCLS; ++c) {
            float v = row[c];
            if (v > m) { m = v; lab = c; }   // strict > keeps first occurrence
        }
        live[tile_base + tid]   = m;
        labels[tile_base + tid] = lab;
    } else {
        int b = tile_base + tid;
        if (b < n) {
            const float* row = scores + (size_t)b * CCLS;
            float m = -3.0e38f; int lab = 0;
            for (int c = 0; c < CCLS; ++c) {
                float v = row[c];
                if (v > m) { m = v; lab = c; }
            }
            live[b] = m;
            labels[b] = lab;
        }
    }
}

// ---------------------------------------------------------------------
// (max, idx) combine with jnp.argmax tie-break (lowest index wins)
// ---------------------------------------------------------------------
__device__ __forceinline__ void argmax_pair(float& m, int& i, float om, int oi)
{
    if (om > m || (om == m && oi < i)) { m = om; i = oi; }
}

// 1024-thread block argmax using wave32 shuffles; result in thread 0.
__device__ __forceinline__ void block_argmax(float& lm, int& li,
                                             float* swm, int* swi)
{
    #pragma unroll
    for (int off = 16; off > 0; off >>= 1) {
        float om = __shfl_xor(lm, off, 32);
        int   oi = __shfl_xor(li, off, 32);
        argmax_pair(lm, li, om, oi);
    }
    const int wid  = threadIdx.x >> 5;
    const int lane = threadIdx.x & 31;
    if (lane == 0) { swm[wid] = lm; swi[wid] = li; }
    __syncthreads();
    if (wid == 0) {
        lm = swm[lane];               // exactly 32 waves in a 1024 block
        li = swi[lane];
        #pragma unroll
        for (int off = 16; off > 0; off >>= 1) {
            float om = __shfl_xor(lm, off, 32);
            int   oi = __shfl_xor(li, off, 32);
            argmax_pair(lm, li, om, oi);
        }
    }
    __syncthreads();                  // make swm/swi reusable
}

// Counter-based grid barrier (counter reset to 0 by hipMemsetAsync each call).
__device__ __forceinline__ void grid_barrier(unsigned* cnt, unsigned target)
{
    __syncthreads();
    if (threadIdx.x == 0) {
        __threadfence();
        atomicAdd(cnt, 1u);
        while (__hip_atomic_load(cnt, __ATOMIC_RELAXED,
                                 __HIP_MEMORY_SCOPE_AGENT) < target) {
            __builtin_amdgcn_s_sleep(1);
        }
        __threadfence();
    }
    __syncthreads();
}

// =====================================================================
// Phase 2: 32-block cooperative greedy NMS, 100 iterations.
// Per iteration: each block's fused (suppress + partial-argmax) sweep,
// wave-shuffle block reduction, double-buffered partial publish, one
// grid barrier, then every block redundantly reduces the 32 partials
// to the identical global winner.
// =====================================================================
__global__ __launch_bounds__(NMS_BLOCK) void nms_phase2_greedy(
    const float* __restrict__ boxes,      // [n,4] xyxy
    float*       __restrict__ live,       // [n] mutable scores
    const int*   __restrict__ labels,     // [n]
    float*       __restrict__ out_boxes,  // [MAX_OUT*4]
    float*       __restrict__ out_scores, // [MAX_OUT]
    float*       __restrict__ out_labels, // [MAX_OUT] (float-encoded)
    float*       __restrict__ part_max,   // [2*NBLK] double-buffered
    int*         __restrict__ part_idx,   // [2*NBLK]
    unsigned*    __restrict__ bar_cnt,    // [1], memset to 0 per call
    int n)
{
    __shared__ float swm[32];
    __shared__ int   swi[32];
    __shared__ float selbox[4];
    __shared__ float s_area;

    const int tid     = threadIdx.x;
    const int bid     = blockIdx.x;
    const int gtid    = bid * NMS_BLOCK + tid;
    const int gstride = NBLK * NMS_BLOCK;
    const float4* __restrict__ b4 = (const float4*)boxes;

    unsigned bar_target = 0;

    // initial per-thread argmax over live[]
    float lm = -3.0e38f; int li = 0x7fffffff;
    for (int e = gtid; e < n; e += gstride) {
        float v = live[e];
        if (v > lm) { lm = v; li = e; }
    }
    block_argmax(lm, li, swm, swi);
    if (tid == 0) { part_max[bid] = lm; part_idx[bid] = li; }   // parity 0
    bar_target += NBLK;
    grid_barrier(bar_cnt, bar_target);

    for (int it = 0; it < MAX_OUT; ++it) {
        const int p = (it & 1) * NBLK;

        // every block redundantly reduces the 32 partials -> same winner
        if (tid == 0) {
            float gm = part_max[p]; int gi = part_idx[p];
            for (int b = 1; b < NBLK; ++b)
                argmax_pair(gm, gi, part_max[p + b], part_idx[p + b]);

            float4 bb = b4[gi];
            selbox[0] = bb.x; selbox[1] = bb.y;
            selbox[2] = bb.z; selbox[3] = bb.w;
            s_area = (bb.z - bb.x) * (bb.w - bb.y);

            if (bid == 0) {
                bool valid = gm > (NEG_INF * 0.5f);
                if (valid) {
                    out_boxes[it * 4 + 0] = bb.x;
                    out_boxes[it * 4 + 1] = bb.y;
                    out_boxes[it * 4 + 2] = bb.z;
                    out_boxes[it * 4 + 3] = bb.w;
                    out_scores[it] = gm;
                    out_labels[it] = (float)labels[gi];
                } else {
                    out_boxes[it * 4 + 0] = 0.0f;
                    out_boxes[it * 4 + 1] = 0.0f;
                    out_boxes[it * 4 + 2] = 0.0f;
                    out_boxes[it * 4 + 3] = 0.0f;
                    out_scores[it] = 0.0f;
                    out_labels[it] = -1.0f;
                }
            }
        }
        __syncthreads();

        const float sx1 = selbox[0], sy1 = selbox[1];
        const float sx2 = selbox[2], sy2 = selbox[3];
        const float sarea = s_area;

        // fused suppression + next-iteration partial argmax over own slice
        lm = -3.0e38f; li = 0x7fffffff;
        for (int e = gtid; e < n; e += gstride) {
            float v = live[e];
            if (v > (NEG_INF * 0.5f)) {          // dead-skip: no box load / IoU
                float4 bb = b4[e];
                float iw = fminf(bb.z, sx2) - fmaxf(bb.x, sx1);
                float ih = fminf(bb.w, sy2) - fmaxf(bb.y, sy1);
                iw = fmaxf(iw, 0.0f);
                ih = fmaxf(ih, 0.0f);
                float inter = iw * ih;
                float area  = (bb.z - bb.x) * (bb.w - bb.y);
                float iou   = inter / (area + sarea - inter + 1e-9f);
                if (iou > IOU_THR) { v = NEG_INF; live[e] = v; }
            }
            if (v > lm) { lm = v; li = e; }
        }

        if (it + 1 < MAX_OUT) {
            block_argmax(lm, li, swm, swi);
            if (tid == 0) {
                part_max[((it + 1) & 1) * NBLK + bid] = lm;
                part_idx[((it + 1) & 1) * NBLK + bid] = li;
            }
            bar_target += NBLK;
            grid_barrier(bar_cnt, bar_target);
        }
    }
}

// =====================================================================
// Host-side launcher
// =====================================================================
extern "C" void kernel_launch(void* const* d_in, const int* in_sizes, int n_in,
                              void* d_out, int out_size, void* d_ws, size_t ws_size,
                              hipStream_t stream)
{
    (void)n_in; (void)out_size; (void)ws_size;

    const float* boxes  = (const float*)d_in[0];   // [1, N, 4] f32
    const float* scores = (const float*)d_in[1];   // [1, N, 80] f32
    const int n = in_sizes[0] / 4;                 // N = 120000

    // workspace layout
    float*    live     = (float*)d_ws;
    int*      labels   = (int*)(live + n);
    float*    part_max = (float*)(labels + n);      // [2*NBLK]
    int*      part_idx = (int*)(part_max + 2 * NBLK);
    unsigned* bar_cnt  = (unsigned*)(part_idx + 2 * NBLK);

    float* out        = (float*)d_out;
    float* out_boxes  = out;                   // [100*4]
    float* out_scores = out + MAX_OUT * 4;     // [100]
    float* out_labels = out + MAX_OUT * 5;     // [100]

    const int grid1 = (n + TILE - 1) / TILE;
    nms_phase1_classmax<<<grid1, TILE, 0, stream>>>(scores, live, labels, n);

    hipMemsetAsync(bar_cnt, 0, sizeof(unsigned), stream);

    nms_phase2_greedy<<<NBLK, NMS_BLOCK, 0, stream>>>(
        boxes, live, labels, out_boxes, out_scores, out_labels,
        part_max, part_idx, bar_cnt, n);
}